// DummyGAT_47725676593415
// MI455X (gfx1250) — compile-verified
//
#include <hip/hip_runtime.h>
#include <stdint.h>

#define D_FEAT 64
#define NEG_SLOPE 0.2f

typedef float v2f __attribute__((ext_vector_type(2)));
typedef float v8f __attribute__((ext_vector_type(8)));

// ---------------------------------------------------------------------------
// Kernel 1: h = x @ W  using V_WMMA_F32_16X16X4_F32 (f32 WMMA, K chained 4x16)
// One wave computes a 16-row x 64-col stripe of h (4 accumulator tiles).
// A layout (16x4 f32): lanes 0-15 -> M=lane, v0/v1 = K{0,1}; lanes 16-31 = K{2,3}
// B layout (4x16 f32): lanes 0-15 -> N=lane, v0/v1 = K{0,1}; lanes 16-31 = K{2,3}
// C/D layout: VGPR r -> M=r (lanes 0-15), M=r+8 (lanes 16-31); N = lane&15
// ---------------------------------------------------------------------------
__global__ void gat_gemm_wmma(const float* __restrict__ x,
                              const float* __restrict__ W,
                              float* __restrict__ h,
                              int n_nodes)
{
    const int wave    = (int)((blockIdx.x * blockDim.x + threadIdx.x) >> 5);
    const int rowBase = wave * 16;
    if (rowBase >= n_nodes) return;            // wave-uniform: EXEC stays all-1s
    const int lane  = threadIdx.x & 31;
    const int laneM = lane & 15;
    const int hi    = lane >> 4;

    v8f acc0 = {}, acc1 = {}, acc2 = {}, acc3 = {};
    const float* xrow = x + (size_t)(rowBase + laneM) * D_FEAT;

    #pragma unroll
    for (int k0 = 0; k0 < D_FEAT; k0 += 4) {
        const int ka = k0 + 2 * hi;
        v2f a;
        a.x = xrow[ka];
        a.y = xrow[ka + 1];
        const float* Wr0 = W + (size_t)ka * D_FEAT;
        const float* Wr1 = Wr0 + D_FEAT;
        v2f b0, b1, b2, b3;
        b0.x = Wr0[laneM];      b0.y = Wr1[laneM];
        b1.x = Wr0[16 + laneM]; b1.y = Wr1[16 + laneM];
        b2.x = Wr0[32 + laneM]; b2.y = Wr1[32 + laneM];
        b3.x = Wr0[48 + laneM]; b3.y = Wr1[48 + laneM];
        acc0 = __builtin_amdgcn_wmma_f32_16x16x4_f32(false, a, false, b0, (short)0, acc0, false, false);
        acc1 = __builtin_amdgcn_wmma_f32_16x16x4_f32(false, a, false, b1, (short)0, acc1, false, false);
        acc2 = __builtin_amdgcn_wmma_f32_16x16x4_f32(false, a, false, b2, (short)0, acc2, false, false);
        acc3 = __builtin_amdgcn_wmma_f32_16x16x4_f32(false, a, false, b3, (short)0, acc3, false, false);
    }

    #pragma unroll
    for (int r = 0; r < 8; ++r) {
        const int row = rowBase + r + 8 * hi;
        float* hrow = h + (size_t)row * D_FEAT;
        hrow[laneM]      = acc0[r];
        hrow[16 + laneM] = acc1[r];
        hrow[32 + laneM] = acc2[r];
        hrow[48 + laneM] = acc3[r];
    }
}

// ---------------------------------------------------------------------------
// Kernel 2: a_src[n] = h[n,:] . att_src ; a_dst[n] = h[n,:] . att_dst
// One wave per node, shuffle reduction over 32 lanes.
// ---------------------------------------------------------------------------
__global__ void gat_att(const float* __restrict__ h,
                        const float* __restrict__ att_src,
                        const float* __restrict__ att_dst,
                        float* __restrict__ a_src,
                        float* __restrict__ a_dst,
                        int n_nodes)
{
    const int node = (int)((blockIdx.x * blockDim.x + threadIdx.x) >> 5);
    if (node >= n_nodes) return;
    const int lane = threadIdx.x & 31;

    const float h0 = h[(size_t)node * D_FEAT + lane];
    const float h1 = h[(size_t)node * D_FEAT + 32 + lane];
    float s = h0 * att_src[lane] + h1 * att_src[32 + lane];
    float d = h0 * att_dst[lane] + h1 * att_dst[32 + lane];

    #pragma unroll
    for (int off = 16; off >= 1; off >>= 1) {
        s += __shfl_xor(s, off, 32);
        d += __shfl_xor(d, off, 32);
    }
    if (lane == 0) {
        a_src[node] = s;
        a_dst[node] = d;
    }
}

// ---------------------------------------------------------------------------
// Kernel 3: single edge pass (max-subtraction algebraically cancels):
//   ex = exp(leakyrelu(a_src[s] + a_dst[t]))
//   accum[t,:] += ex * h[s,:]   (global_atomic_add_f32, L2-resident)
//   denom[t]   += ex
// One wave per edge; edges [E, E+N) are the implicit self loops.
// ---------------------------------------------------------------------------
__global__ void gat_edge(const long long* __restrict__ ei,   // [2, E] int64
                         const float* __restrict__ a_src,
                         const float* __restrict__ a_dst,
                         const float* __restrict__ h,
                         float* __restrict__ accum,
                         float* __restrict__ denom,
                         int n_edges, int n_nodes)
{
    const int w = (int)((blockIdx.x * blockDim.x + threadIdx.x) >> 5);
    const int total = n_edges + n_nodes;
    if (w >= total) return;
    const int lane = threadIdx.x & 31;

    int s, t;
    if (w < n_edges) {
        s = (int)ei[w];
        t = (int)ei[(size_t)n_edges + w];
    } else {
        s = t = w - n_edges;      // self loop
    }

    float e = a_src[s] + a_dst[t];
    e = (e >= 0.f) ? e : NEG_SLOPE * e;
    const float ex = __expf(e);

    const float2 hv = ((const float2*)(h + (size_t)s * D_FEAT))[lane]; // 256B coalesced row
    float* arow = accum + (size_t)t * D_FEAT;
    atomicAdd(arow + 2 * lane,     ex * hv.x);
    atomicAdd(arow + 2 * lane + 1, ex * hv.y);
    if (lane == 0) atomicAdd(denom + t, ex);
}

// ---------------------------------------------------------------------------
// Kernel 4: out[n,d] = accum[n,d] / denom[n] + bias[d]
// ---------------------------------------------------------------------------
__global__ void gat_finalize(const float* __restrict__ accum,
                             const float* __restrict__ denom,
                             const float* __restrict__ bias,
                             float* __restrict__ out,
                             int n_nodes)
{
    const size_t i = (size_t)blockIdx.x * blockDim.x + threadIdx.x;
    if (i >= (size_t)n_nodes * D_FEAT) return;
    const int n = (int)(i >> 6);
    const int d = (int)(i & 63);
    out[i] = accum[i] / denom[n] + bias[d];
}

// ---------------------------------------------------------------------------
extern "C" void kernel_launch(void* const* d_in, const int* in_sizes, int n_in,
                              void* d_out, int out_size, void* d_ws, size_t ws_size,
                              hipStream_t stream)
{
    const float*     x        = (const float*)d_in[0];
    const long long* ei       = (const long long*)d_in[1];   // int64 edge_index
    const float*     W        = (const float*)d_in[2];
    const float*     att_src  = (const float*)d_in[3];
    const float*     att_dst  = (const float*)d_in[4];
    const float*     bias     = (const float*)d_in[5];
    float*           out      = (float*)d_out;

    const int n_nodes = in_sizes[0] / D_FEAT;
    const int n_edges = in_sizes[1] / 2;

    // workspace layout (floats): h | a_src | a_dst | denom | accum
    float* ws      = (float*)d_ws;
    float* h       = ws;
    float* a_src   = h + (size_t)n_nodes * D_FEAT;
    float* a_dst   = a_src + n_nodes;
    float* denom   = a_dst + n_nodes;
    float* accum   = denom + n_nodes;   // adjacent to denom: zero both in one memset

    // zero denom + accum (graph-capturable async memset)
    hipMemsetAsync(denom, 0, sizeof(float) * ((size_t)n_nodes * (D_FEAT + 1)), stream);

    const int BLK = 256;

    // 1) h = x @ W via f32 WMMA: one wave per 16-row stripe
    {
        const int waves   = (n_nodes + 15) / 16;
        const long long threads = (long long)waves * 32;
        const int blocks  = (int)((threads + BLK - 1) / BLK);
        gat_gemm_wmma<<<blocks, BLK, 0, stream>>>(x, W, h, n_nodes);
    }

    // 2) attention coefficients: one wave per node
    {
        const long long threads = (long long)n_nodes * 32;
        const int blocks = (int)((threads + BLK - 1) / BLK);
        gat_att<<<blocks, BLK, 0, stream>>>(h, att_src, att_dst, a_src, a_dst, n_nodes);
    }

    // 3) fused edge pass (exp + weighted scatter-add): one wave per edge
    {
        const long long total   = (long long)n_edges + n_nodes;
        const long long threads = total * 32;
        const int blocks = (int)((threads + BLK - 1) / BLK);
        gat_edge<<<blocks, BLK, 0, stream>>>(ei, a_src, a_dst, h, accum, denom,
                                             n_edges, n_nodes);
    }

    // 4) normalize + bias
    {
        const long long threads = (long long)n_nodes * D_FEAT;
        const int blocks = (int)((threads + BLK - 1) / BLK);
        gat_finalize<<<blocks, BLK, 0, stream>>>(accum, denom, bias, out, n_nodes);
    }
}